// RegularizedFMNet_30047591203015
// MI455X (gfx1250) — compile-verified
//
#include <hip/hip_runtime.h>

// RegularizedFMNet for MI455X (gfx1250, wave32).
//  Inputs (dict order): A [8,200,352] f32, B [8,200,352] f32,
//                       evals_x [8,200] f32, evals_y [8,200] f32.
//  Output: Cxy [8,200,200] f32.
//
// Pipeline:
//  1) prep_kernel : per-batch scale = max(evals), e1/e2 = sqrt(evals/scale) (gamma=0.5)
//  2) gemm_kernel : AAt = A·Aᵀ and BAt = B·Aᵀ fused in one pass via
//                   V_WMMA_F32_16X16X4_F32; one wave per (ti,tj) produces both 16x16
//                   tiles (shared N-side fragment), 88 K-steps over C=352.
//                   Out-of-range lanes are CLAMPED (not predicated): they only write
//                   the 208-pad region, which is never read -> branch-free inner loop.
//  3) solve_kernel: 1600 workgroups (one per (b,i)); 200x200 fp32 system resident in
//                   LDS (stride 202 -> conflict-free b64 rows), pivot-free Gaussian
//                   elimination (SPD), v2f-vectorized rank-1 updates, in-place
//                   back-substitution.

typedef float v2f __attribute__((ext_vector_type(2)));
typedef float v4f __attribute__((ext_vector_type(4)));
typedef float v8f __attribute__((ext_vector_type(8)));

#define BSZ   8
#define KDIM  200
#define CDIM  352
#define LDP   208      // padded leading dim for AAt/BAt (13*16)
#define NT    13       // 16-wide tiles covering 200
#define LMBDA 0.01f
#define MSTR  202      // LDS stride (dwords): even -> 8B-aligned rows; 202%64=10,
                       // gcd(10,64)=2 -> 32 lanes hit 32 distinct even banks (b64-safe)

// ---------------- prep: scale + e1/e2 ----------------
__global__ __launch_bounds__(256)
void fmnet_prep_kernel(const float* __restrict__ ex, const float* __restrict__ ey,
                       float* __restrict__ e1, float* __restrict__ e2) {
  __shared__ float red[256];
  const int b = blockIdx.x, t = threadIdx.x;
  float m = 0.0f;
  for (int k = t; k < KDIM; k += 256)
    m = fmaxf(m, fmaxf(ex[b * KDIM + k], ey[b * KDIM + k]));
  red[t] = m;
  __syncthreads();
  for (int s = 128; s > 0; s >>= 1) {
    if (t < s) red[t] = fmaxf(red[t], red[t + s]);
    __syncthreads();
  }
  const float inv = 1.0f / red[0];
  for (int k = t; k < KDIM; k += 256) {
    e1[b * KDIM + k] = sqrtf(ex[b * KDIM + k] * inv);  // (x/scale)^0.5
    e2[b * KDIM + k] = sqrtf(ey[b * KDIM + k] * inv);
  }
}

// ---------------- fused fp32 WMMA GEMM: AAt = A·Aᵀ, BAt = B·Aᵀ ----------------
// grid: x = 169 tile pairs, y = batch. One wave -> one 16x16 tile of each output.
__global__ __launch_bounds__(32)
void fmnet_gemm_kernel(const float* __restrict__ A, const float* __restrict__ Bm,
                       float* __restrict__ AAt, float* __restrict__ BAt) {
  const int ti   = blockIdx.x % NT;
  const int tj   = blockIdx.x / NT;
  const int b    = blockIdx.y;
  const int lane = threadIdx.x;          // 0..31, wave32

  const int l15  = lane & 15;
  const int ksel = (lane >> 4) * 2;      // lanes 16..31 hold K+2,K+3 (ISA 7.12.2)
  // Clamp instead of predicate: tiles at ti/tj==12 duplicate row 199 into the
  // pad region of the 208-padded outputs (never read by the solver).
  const int mrow = min(ti * 16 + l15, KDIM - 1);   // M side (rows of A or B)
  const int ncol = min(tj * 16 + l15, KDIM - 1);   // N side (rows of A, transposed)

  const float* arowA = A  + ((size_t)b * KDIM + mrow) * CDIM + ksel;
  const float* arowB = Bm + ((size_t)b * KDIM + mrow) * CDIM + ksel;
  const float* brow  = A  + ((size_t)b * KDIM + ncol) * CDIM + ksel;

  v8f accA = {}, accB = {};
#pragma unroll 4
  for (int k0 = 0; k0 < CDIM; k0 += 4) {
    const v2f bf = *(const v2f*)(brow  + k0);   // shared N-side fragment
    const v2f aA = *(const v2f*)(arowA + k0);
    const v2f aB = *(const v2f*)(arowB + k0);
    accA = __builtin_amdgcn_wmma_f32_16x16x4_f32(false, aA, false, bf,
                                                 (short)0, accA, false, false);
    accB = __builtin_amdgcn_wmma_f32_16x16x4_f32(false, aB, false, bf,
                                                 (short)0, accB, false, false);
  }

  // C/D layout: VGPR v -> M = v (lanes 0-15) or v+8 (lanes 16-31); N = lane&15.
  const size_t obase = (size_t)b * LDP * LDP;
  const int rbase = ti * 16 + (lane >> 4) * 8;
  const int ccol  = tj * 16 + l15;
#pragma unroll
  for (int v = 0; v < 8; ++v) {
    AAt[obase + (size_t)(rbase + v) * LDP + ccol] = accA[v];
    BAt[obase + (size_t)(rbase + v) * LDP + ccol] = accB[v];
  }
}

// ---------------- per-(b,i) SPD solve in LDS ----------------
// grid: x = i in 0..199, y = b in 0..7; 256 threads (8 waves).
__global__ __launch_bounds__(256)
void fmnet_solve_kernel(const float* __restrict__ AAt, const float* __restrict__ BAt,
                        const float* __restrict__ e1, const float* __restrict__ e2,
                        float* __restrict__ out) {
  __shared__ float M[KDIM * MSTR];   // 161,600 B
  __shared__ float rhs[KDIM];        // +800 B  -> 162.4 KB total

  const int i = blockIdx.x, b = blockIdx.y, t = threadIdx.x;
  const float* S = AAt + (size_t)b * LDP * LDP;

  // Load Gram matrix into LDS: b128 global loads, 2x b64 LDS stores per chunk.
  for (int idx = t; idx < KDIM * (KDIM / 4); idx += 256) {
    const int r = idx / (KDIM / 4);
    const int c = (idx - r * (KDIM / 4)) * 4;
    const v4f q = *(const v4f*)(S + (size_t)r * LDP + c);        // 16B aligned
    *(v2f*)(&M[r * MSTR + c])     = v2f{q.x, q.y};               // 8B aligned
    *(v2f*)(&M[r * MSTR + c + 2]) = v2f{q.z, q.w};
  }
  // rhs = row i of B·Aᵀ ; diagonal += lambda * D[b,i,r] computed on the fly.
  const float e2i = e2[b * KDIM + i];
  for (int r = t; r < KDIM; r += 256) {
    rhs[r] = BAt[(size_t)b * LDP * LDP + (size_t)i * LDP + r];
    const float e1r   = e1[b * KDIM + r];
    const float denom = e1r * e1r + e2i * e2i;
    const float mre   = e2i - e1r;
    const float d     = (mre * mre + 1.0f) / (denom * denom);
    M[r * MSTR + r] += LMBDA * d;
  }

  // Forward elimination (no pivoting; matrix is SPD). Rows are thread-parallel,
  // columns run vectorized (v2f) per thread.
  for (int p = 0; p < KDIM - 1; ++p) {
    __syncthreads();
    const float pivinv = 1.0f / M[p * MSTR + p];   // LDS broadcast
    const float rp     = rhs[p];
    const float* __restrict__ rowp = &M[p * MSTR];
    for (int r = p + 1 + t; r < KDIM; r += 256) {
      float* __restrict__ rowr = &M[r * MSTR];
      const float f = rowr[p] * pivinv;
      int c = p + 1;
      if (c & 1) { rowr[c] -= f * rowp[c]; ++c; }  // align to 8B
      for (; c < KDIM; c += 2) {                   // KDIM even -> no tail
        const v2f pv = *(const v2f*)(rowp + c);
        v2f rv = *(v2f*)(rowr + c);
        rv.x -= f * pv.x;
        rv.y -= f * pv.y;
        *(v2f*)(rowr + c) = rv;
      }
      rhs[r] -= f * rp;
    }
  }

  // In-place back-substitution: rhs becomes x.
  for (int p = KDIM - 1; p >= 0; --p) {
    __syncthreads();
    if (t == 0) rhs[p] = rhs[p] / M[p * MSTR + p];
    __syncthreads();
    const float xp = rhs[p];
    for (int r = t; r < p; r += 256)
      rhs[r] -= M[r * MSTR + p] * xp;
  }
  __syncthreads();

  for (int c = t; c < KDIM; c += 256)
    out[((size_t)b * KDIM + i) * KDIM + c] = rhs[c];
}

extern "C" void kernel_launch(void* const* d_in, const int* in_sizes, int n_in,
                              void* d_out, int out_size, void* d_ws, size_t ws_size,
                              hipStream_t stream) {
  const float* A  = (const float*)d_in[0];
  const float* Bm = (const float*)d_in[1];
  const float* ex = (const float*)d_in[2];
  const float* ey = (const float*)d_in[3];
  float* out = (float*)d_out;

  // Workspace layout (floats): AAt | BAt | e1 | e2   (~2.78 MB total)
  float* ws  = (float*)d_ws;
  const size_t padMat = (size_t)BSZ * LDP * LDP;   // 346,112
  float* AAt = ws;
  float* BAt = ws + padMat;
  float* e1  = ws + 2 * padMat;
  float* e2  = ws + 2 * padMat + (size_t)BSZ * KDIM;

  fmnet_prep_kernel<<<dim3(BSZ), dim3(256), 0, stream>>>(ex, ey, e1, e2);
  fmnet_gemm_kernel<<<dim3(NT * NT, BSZ), dim3(32), 0, stream>>>(A, Bm, AAt, BAt);
  fmnet_solve_kernel<<<dim3(KDIM, BSZ), dim3(256), 0, stream>>>(AAt, BAt, e1, e2, out);
}